// AffineImageTransformation_19542101197242
// MI455X (gfx1250) — compile-verified
//
#include <hip/hip_runtime.h>

typedef float v4f __attribute__((ext_vector_type(4)));
typedef float v2f __attribute__((ext_vector_type(2)));

// B=32, C=3, H=W=512 input; 512x512 output. f32 in/out.
// One thread computes 4 consecutive output-x pixels for all 3 channels.
// Block = 256 threads -> 1024 pixels, 256 blocks per image, 8192 blocks total.
// x-corner pairs are fetched with one (unaligned) b64 gather per row per
// channel; clamp collisions at the image edges are resolved with v_cndmask,
// and every case where the selected value is wrong has bilinear weight == 0.
__global__ __launch_bounds__(256) void affine_bilinear_kernel(
    const float* __restrict__ img,    // (32, 3, 512, 512)
    const float* __restrict__ theta,  // (32, 2, 3)
    float* __restrict__ out) {        // (32, 3, 512, 512)
  constexpr int W = 512, H = 512;
  constexpr int PLANE = W * H;

  const int blk = blockIdx.x;
  const int b = blk >> 8;                                   // image index (uniform)
  const int pixBase = ((blk & 255) << 10) | (threadIdx.x << 2);
  const int y  = pixBase >> 9;                              // output row
  const int xp = pixBase & (W - 1);                         // output col (multiple of 4)

  // theta is block-uniform -> lowers to scalar s_load into SGPRs.
  const float* t = theta + b * 6;
  const float t00 = t[0], t01 = t[1], t02 = t[2];
  const float t10 = t[3], t11 = t[4], t12 = t[5];

  // align_corners=False base coords: s = (2*p + 1)/N - 1
  const float ysn = (2.0f * (float)y  + 1.0f) * (1.0f / (float)H) - 1.0f;
  const float xsn = (2.0f * (float)xp + 1.0f) * (1.0f / (float)W) - 1.0f;

  // grid coords, then unnormalize: i = ((g + 1)*N - 1) * 0.5 = g*N/2 + (N/2 - 0.5)
  const float gx = fmaf(t00, xsn, fmaf(t01, ysn, t02));
  const float gy = fmaf(t10, xsn, fmaf(t11, ysn, t12));
  const float ix0 = fmaf(gx, 0.5f * (float)W, 0.5f * (float)W - 0.5f);
  const float iy0 = fmaf(gy, 0.5f * (float)H, 0.5f * (float)H - 0.5f);
  // Per +1 output pixel in x: xsn steps 2/W, so ix steps t00 and iy steps t10.

  const float* __restrict__ plane0 = img + (size_t)b * 3 * PLANE;

  v4f r0, r1, r2;

#pragma unroll
  for (int j = 0; j < 4; ++j) {
    const float jx = fmaf((float)j, t00, ix0);
    const float jy = fmaf((float)j, t10, iy0);

    const float fx = floorf(jx);
    const float fy = floorf(jy);
    const float wx1 = jx - fx, wx0 = 1.0f - wx1;
    const float wy1 = jy - fy, wy0 = 1.0f - wy1;

    const int x0 = (int)fx, y0 = (int)fy;
    const int x1 = x0 + 1,  y1 = y0 + 1;

    const bool vx0 = (unsigned)x0 < (unsigned)W;
    const bool vx1 = (unsigned)x1 < (unsigned)W;
    const bool vy0 = (unsigned)y0 < (unsigned)H;
    const bool vy1 = (unsigned)y1 < (unsigned)H;

    // zeros padding folded into the weights (branch-free gathers below)
    const float m00 = (vx0 && vy0) ? wx0 * wy0 : 0.0f;
    const float m10 = (vx1 && vy0) ? wx1 * wy0 : 0.0f;
    const float m01 = (vx0 && vy1) ? wx0 * wy1 : 0.0f;
    const float m11 = (vx1 && vy1) ? wx1 * wy1 : 0.0f;

    // One b64 gather covers both x-corners of a row.
    const int xl  = min(max(x0, 0), W - 2);        // pair base column
    const int cy0 = min(max(y0, 0), H - 1);
    const int cy1 = min(max(y1, 0), H - 1);
    const int oT = cy0 * W + xl;                   // top-row pair offset
    const int oB = cy1 * W + xl;                   // bottom-row pair offset

    // Clamp-collision selectors (uniform across channels/rows):
    const bool useHi0 = (x0 >= W - 1);  // corner0 clamped onto pair.y (right edge)
    const bool useLo1 = (x0 < 0);       // corner1 clamped onto pair.x (left edge)

#pragma unroll
    for (int c = 0; c < 3; ++c) {
      const float* __restrict__ p = plane0 + c * PLANE;

      v2f pT, pB;                                  // {x0, x0+1} pairs, 4B-aligned
      __builtin_memcpy(&pT, p + oT, sizeof(v2f));
      __builtin_memcpy(&pB, p + oB, sizeof(v2f));

      const float vT0 = useHi0 ? pT.y : pT.x;
      const float vT1 = useLo1 ? pT.x : pT.y;
      const float vB0 = useHi0 ? pB.y : pB.x;
      const float vB1 = useLo1 ? pB.x : pB.y;

      float v = m00 * vT0;
      v = fmaf(m10, vT1, v);
      v = fmaf(m01, vB0, v);
      v = fmaf(m11, vB1, v);
      if (c == 0)      r0[j] = v;
      else if (c == 1) r1[j] = v;
      else             r2[j] = v;
    }
  }

  // Output is write-once: non-temporal b128 stores keep the gathered input
  // resident in the 192MB L2 instead of being evicted by the output stream.
  float* __restrict__ op = out + (size_t)b * 3 * PLANE + y * W + xp;
  __builtin_nontemporal_store(r0, (v4f*)(op));
  __builtin_nontemporal_store(r1, (v4f*)(op + PLANE));
  __builtin_nontemporal_store(r2, (v4f*)(op + 2 * PLANE));
}

extern "C" void kernel_launch(void* const* d_in, const int* in_sizes, int n_in,
                              void* d_out, int out_size, void* d_ws, size_t ws_size,
                              hipStream_t stream) {
  const float* img   = (const float*)d_in[0];   // 32*3*512*512 f32
  const float* theta = (const float*)d_in[1];   // 32*2*3 f32
  float* out = (float*)d_out;                   // 32*3*512*512 f32

  // 32 images * 256 blocks/image
  dim3 grid(32 * 256);
  dim3 block(256);
  affine_bilinear_kernel<<<grid, block, 0, stream>>>(img, theta, out);
}